// GCN4_58128087384868
// MI455X (gfx1250) — compile-verified
//
#include <hip/hip_runtime.h>
#include <hip/hip_bf16.h>

typedef __attribute__((ext_vector_type(2))) float v2f;
typedef __attribute__((ext_vector_type(8))) float v8f;

// ---------------------------------------------------------------------------
// Dense GEMM  C[M x NCOLS] = A[M x K] * W[K x NCOLS]  via V_WMMA_F32_16X16X4_F32
// Block: 256 threads = 8 waves; each wave computes one 16x16 tile.
// Block footprint: 32 rows x 64 cols.
// A 16x4 f32 fragment (ISA 7.12.2): lanes 0-15 -> rows, VGPR0/1 = K,K+1;
// lanes 16-31 -> same rows, VGPR0/1 = K+2,K+3.
// B 4x16 f32 fragment: VGPR0 = rows K (lanes 0-15) / K+2 (lanes 16-31),
// VGPR1 = rows K+1 / K+3 (mirrors the C/D half-split layout).
// C/D 16x16 f32: acc[r] at lane L -> row base_m + r + (L>=16 ? 8 : 0), col = L&15.
// ---------------------------------------------------------------------------
template <int K, int NCOLS>
__global__ __launch_bounds__(256) void gcn_gemm_wmma(const float* __restrict__ A,
                                                     const float* __restrict__ W,
                                                     float* __restrict__ C, int M) {
    constexpr int NPAD = (NCOLS + 15) & ~15;
    const int wave = threadIdx.x >> 5;
    const int lane = threadIdx.x & 31;
    const int wx = wave & 3;   // column tile 0..3
    const int wy = wave >> 2;  // row tile 0..1
    const int base_n = wx * 16;
    if (base_n >= NPAD) return;  // wave-uniform: EXEC stays all-ones elsewhere

    const int base_m = blockIdx.x * 32 + wy * 16;
    const int hi  = lane >> 4;   // 0: lanes 0-15, 1: lanes 16-31
    const int l15 = lane & 15;
    const int col = base_n + l15;
    // col-guard is statically dead when NCOLS is a multiple of 16
    const bool colOK = ((NCOLS & 15) == 0) ? true : (col < NCOLS);

    int arow = base_m + l15;
    if (arow >= M) arow = M - 1;  // clamp; edge rows masked on store
    const float* __restrict__ ap = A + (size_t)arow * K;

    v8f acc = {};
#pragma unroll 4
    for (int k = 0; k < K; k += 4) {
        const int kk = k + hi * 2;
        v2f a, b;
        a.x = ap[kk + 0];
        a.y = ap[kk + 1];
        float b0 = 0.0f, b1 = 0.0f;
        if (colOK) {
            b0 = W[(size_t)(kk + 0) * NCOLS + col];
            b1 = W[(size_t)(kk + 1) * NCOLS + col];
        }
        b.x = b0;
        b.y = b1;
        // (neg_a, A, neg_b, B, c_mod, C, reuse_a, reuse_b)
        acc = __builtin_amdgcn_wmma_f32_16x16x4_f32(false, a, false, b,
                                                    (short)0, acc, false, false);
    }

    // Epilogue: wave-uniform full-tile fast path -> 8 straight stores with
    // immediate offsets off one base pointer; guarded path only on the ragged
    // final row-block.
    float* __restrict__ cp = C + (size_t)(base_m + hi * 8) * NCOLS + col;
    if (base_m + 16 <= M) {          // wave-uniform (s_cbranch, no exec churn)
        if (colOK) {
#pragma unroll
            for (int r = 0; r < 8; ++r) cp[(size_t)r * NCOLS] = acc[r];
        }
    } else {
#pragma unroll
        for (int r = 0; r < 8; ++r) {
            const int row = base_m + hi * 8 + r;
            if (colOK && row < M) cp[(size_t)r * NCOLS] = acc[r];
        }
    }
}

// ---------------------------------------------------------------------------
// SpMM scatter:  agg[dst[e], f] += sup[src[e], f] * w[e]
// One thread per (edge, feature). For D = 64 the lanes of a wave share an edge, so the
// edge-index loads are wave-uniform; the float atomics hit the 192 MB L2
// (support/agg buffers are fully L2-resident).
// ---------------------------------------------------------------------------
template <int D>
__global__ __launch_bounds__(256) void gcn_spmm(const float* __restrict__ sup,
                                                const int* __restrict__ src,
                                                const int* __restrict__ dst,
                                                const float* __restrict__ w,
                                                float* __restrict__ agg, int nE) {
    const long long i = (long long)blockIdx.x * blockDim.x + threadIdx.x;
    const long long total = (long long)nE * D;
    if (i >= total) return;
    const int e = (int)(i / D);
    const int f = (int)(i - (long long)e * D);
    const float v = sup[(size_t)src[e] * D + f] * w[e];
    unsafeAtomicAdd(&agg[(size_t)dst[e] * D + f], v);
}

// ---------------------------------------------------------------------------
// In-place bias + ReLU over [M x D] (D = 64 here).
// ---------------------------------------------------------------------------
template <int D>
__global__ __launch_bounds__(256) void gcn_bias_relu(float* __restrict__ h,
                                                     const float* __restrict__ b,
                                                     long long total) {
    const long long i = (long long)blockIdx.x * blockDim.x + threadIdx.x;
    if (i >= total) return;
    const int f = (int)(i & (D - 1));  // D power of two
    const float v = h[i] + b[f];
    h[i] = v > 0.0f ? v : 0.0f;
}

// ---------------------------------------------------------------------------
// Fused bias + log_softmax over rows of 40 classes. One wave per row; lane l
// covers classes l and l+32; wave32 shuffle reductions for max and sum.
// ---------------------------------------------------------------------------
__global__ __launch_bounds__(256) void gcn_logsoftmax(const float* __restrict__ in,
                                                      const float* __restrict__ b,
                                                      float* __restrict__ out, int M) {
    const int row  = blockIdx.x * 8 + (threadIdx.x >> 5);
    const int lane = threadIdx.x & 31;
    if (row >= M) return;
    const float* r = in + (size_t)row * 40;

    const float x0 = r[lane] + b[lane];  // lanes 0..31 all < 40
    const bool  has1 = (lane + 32) < 40;
    const float x1 = has1 ? (r[lane + 32] + b[lane + 32]) : -3.402823466e38f;

    float m = fmaxf(x0, x1);
#pragma unroll
    for (int off = 16; off; off >>= 1) m = fmaxf(m, __shfl_xor(m, off, 32));

    float s = __expf(x0 - m) + (has1 ? __expf(x1 - m) : 0.0f);
#pragma unroll
    for (int off = 16; off; off >>= 1) s += __shfl_xor(s, off, 32);

    const float lg = m + __logf(s);
    out[(size_t)row * 40 + lane] = x0 - lg;
    if (has1) out[(size_t)row * 40 + lane + 32] = x1 - lg;
}

// ---------------------------------------------------------------------------
// Orchestration. Two ping-pong workspace buffers:
//   sup  = d_ws[0 .. M*64)           (GEMM output / SpMM input)
//   agg  = d_ws[M*64 .. 2*M*64)      (SpMM accumulator -> ReLU'd h in place)
// ---------------------------------------------------------------------------
extern "C" void kernel_launch(void* const* d_in, const int* in_sizes, int n_in,
                              void* d_out, int out_size, void* d_ws, size_t ws_size,
                              hipStream_t stream) {
    const float* x    = (const float*)d_in[0];
    const int*   esrc = (const int*)d_in[1];
    const int*   edst = (const int*)d_in[2];
    const float* ew   = (const float*)d_in[3];
    const float* W1   = (const float*)d_in[4];
    const float* b1   = (const float*)d_in[5];
    const float* W2   = (const float*)d_in[6];
    const float* b2   = (const float*)d_in[7];
    const float* W3   = (const float*)d_in[8];
    const float* b3   = (const float*)d_in[9];
    const float* W4   = (const float*)d_in[10];
    const float* b4   = (const float*)d_in[11];

    const int M = in_sizes[0] / 128;
    const int E = in_sizes[1];

    float* sup = (float*)d_ws;
    float* agg = sup + (size_t)M * 64;

    const int gemm_blocks = (M + 31) / 32;
    const long long tot64 = (long long)M * 64;
    const int ew_blocks   = (int)((tot64 + 255) / 256);
    const long long se64  = (long long)E * 64;
    const int sp64_blocks = (int)((se64 + 255) / 256);
    const long long se40  = (long long)E * 40;
    const int sp40_blocks = (int)((se40 + 255) / 256);

    // ---- Layer 1: x[M,128] @ W1[128,64] -> spmm -> relu(+b1) ----
    gcn_gemm_wmma<128, 64><<<gemm_blocks, 256, 0, stream>>>(x, W1, sup, M);
    hipMemsetAsync(agg, 0, (size_t)M * 64 * sizeof(float), stream);
    gcn_spmm<64><<<sp64_blocks, 256, 0, stream>>>(sup, esrc, edst, ew, agg, E);
    gcn_bias_relu<64><<<ew_blocks, 256, 0, stream>>>(agg, b1, tot64);

    // ---- Layer 2 ----
    gcn_gemm_wmma<64, 64><<<gemm_blocks, 256, 0, stream>>>(agg, W2, sup, M);
    hipMemsetAsync(agg, 0, (size_t)M * 64 * sizeof(float), stream);
    gcn_spmm<64><<<sp64_blocks, 256, 0, stream>>>(sup, esrc, edst, ew, agg, E);
    gcn_bias_relu<64><<<ew_blocks, 256, 0, stream>>>(agg, b2, tot64);

    // ---- Layer 3 ----
    gcn_gemm_wmma<64, 64><<<gemm_blocks, 256, 0, stream>>>(agg, W3, sup, M);
    hipMemsetAsync(agg, 0, (size_t)M * 64 * sizeof(float), stream);
    gcn_spmm<64><<<sp64_blocks, 256, 0, stream>>>(sup, esrc, edst, ew, agg, E);
    gcn_bias_relu<64><<<ew_blocks, 256, 0, stream>>>(agg, b3, tot64);

    // ---- Layer 4: [M,64] @ W4[64,40] -> spmm -> bias + log_softmax ----
    gcn_gemm_wmma<64, 40><<<gemm_blocks, 256, 0, stream>>>(agg, W4, sup, M);
    hipMemsetAsync(agg, 0, (size_t)M * 40 * sizeof(float), stream);
    gcn_spmm<40><<<sp40_blocks, 256, 0, stream>>>(sup, esrc, edst, ew, agg, E);
    gcn_logsoftmax<<<(M + 7) / 8, 256, 0, stream>>>(agg, b4, (float*)d_out, M);
}